// SpatialLoss_377957122722
// MI455X (gfx1250) — compile-verified
//
#include <hip/hip_runtime.h>

typedef __attribute__((ext_vector_type(2))) float v2f;
typedef __attribute__((ext_vector_type(8))) float v8f;

#define B_IMG 16
#define H_IN 512
#define W_IN 512
#define HP 128
#define WP 128
#define NE (B_IMG * HP * WP)      /* 262144 pooled elements */
#define INV_POOL (1.0f / 48.0f)   /* channel mean (1/3) * 4x4 pool mean (1/16) */

// ---------------------------------------------------------------------------
// Phase 1: m[b,i,j] = (1/48) * sum over 3 channels x 4x4 window of (pred-target)
// One thread per pooled element; 24 coalesced float4 loads/thread; each input
// byte is read exactly once -> HBM-bandwidth-bound (~100.7MB / 23.3TB/s ~ 4.3us).
// ---------------------------------------------------------------------------
__global__ __launch_bounds__(256) void pool_diff_kernel(
    const float* __restrict__ pred, const float* __restrict__ targ,
    float* __restrict__ m)
{
    int idx = blockIdx.x * blockDim.x + threadIdx.x;
    if (idx >= NE) return;
    int j = idx & (WP - 1);
    int i = (idx >> 7) & (HP - 1);
    int b = idx >> 14;

    float acc = 0.f;
#pragma unroll
    for (int ch = 0; ch < 3; ++ch) {
#pragma unroll
        for (int r = 0; r < 4; ++r) {
            int rowbase = ((b * 3 + ch) * H_IN + (i * 4 + r)) * W_IN;
            const float4* p4 = (const float4*)(pred + rowbase);
            const float4* t4 = (const float4*)(targ + rowbase);
            float4 p = p4[j];   // 16B-aligned, consecutive lanes -> contiguous
            float4 t = t4[j];
            acc += (p.x - t.x) + (p.y - t.y) + (p.z - t.z) + (p.w - t.w);
        }
    }
    m[idx] = acc * INV_POOL;
}

// ---------------------------------------------------------------------------
// Phase 2: E = sum of 4 directional squared diffs of m (zero-padded), reduced
// with V_WMMA_F32_16X16X4_F32: C += A x ones(4x16). With B = all-ones,
// D[mm,n] = sum_k A[mm,k] + C[mm,n], so any packing of 64 E-values into A
// yields the chunk total in the N=0 column of C (lanes 0 and 16, VGPRs 0..7).
// Boundary handling is branchless (clamped addresses + value select) so the 5
// neighbor loads issue unconditionally and EXEC stays all-1s for the WMMA.
// INVARIANT: grid must satisfy (gridDim.x * wavesPerBlk * 64) divides NE.
// ---------------------------------------------------------------------------
__global__ __launch_bounds__(256) void grad_energy_wmma_kernel(
    const float* __restrict__ m, float* __restrict__ partials)
{
    const int lane        = threadIdx.x & 31;
    const int waveInBlk   = threadIdx.x >> 5;
    const int wavesPerBlk = blockDim.x >> 5;                 // 8
    const int gwave       = blockIdx.x * wavesPerBlk + waveInBlk;
    const int totWaves    = gridDim.x * wavesPerBlk;         // 2048

    v8f c = {};                               // fp32 accumulator fragment
    v2f ones; ones.x = 1.0f; ones.y = 1.0f;   // B = all-ones 4x16

    for (int base = gwave * 64; base < NE; base += totWaves * 64) {
        v2f a;
#pragma unroll
        for (int t = 0; t < 2; ++t) {
            int idx = base + lane * 2 + t;    // always < NE by grid invariant
            int j = idx & 127;
            int r = (idx >> 7) & 127;
            bool hasL = (j > 0), hasR = (j < 127);
            bool hasU = (r > 0), hasD = (r < 127);
            // Clamped, unconditional loads (all in-bounds), then mask values.
            float ce = m[idx];
            float l  = m[idx - (hasL ? 1   : 0)];
            float rt = m[idx + (hasR ? 1   : 0)];
            float u  = m[idx - (hasU ? 128 : 0)];
            float d  = m[idx + (hasD ? 128 : 0)];
            l  = hasL ? l  : 0.f;             // zero padding at the borders
            rt = hasR ? rt : 0.f;
            u  = hasU ? u  : 0.f;
            d  = hasD ? d  : 0.f;
            float dl = ce - l, dr = ce - rt, du = ce - u, dd = ce - d;
            float e  = dl * dl + dr * dr + du * du + dd * dd;
            if (t == 0) a.x = e; else a.y = e;
        }
        // 8 args: (neg_a, A, neg_b, B, c_mod, C, reuse_a, reuse_b)
        c = __builtin_amdgcn_wmma_f32_16x16x4_f32(
            false, a, false, ones, (short)0, c, false, false);
    }

    // Per-lane sum of the 8 C VGPRs; the true chunk total is the N=0 column:
    // lane 0 carries rows M=0..7, lane 16 carries rows M=8..15.
    float s = c[0] + c[1] + c[2] + c[3] + c[4] + c[5] + c[6] + c[7];
    float waveSum = __shfl(s, 0, 32) + __shfl(s, 16, 32);

    __shared__ float ws[32];
    if (lane == 0) ws[waveInBlk] = waveSum;
    __syncthreads();
    if (threadIdx.x == 0) {
        float t = 0.f;
        for (int w = 0; w < wavesPerBlk; ++w) t += ws[w];
        partials[blockIdx.x] = t;   // deterministic: no float atomics
    }
}

// ---------------------------------------------------------------------------
// Phase 3: reduce 256 block partials, scale by 1/NE (LOSS_WEIGHT == 1).
// ---------------------------------------------------------------------------
__global__ __launch_bounds__(256) void finalize_kernel(
    const float* __restrict__ partials, float* __restrict__ out, int n)
{
    __shared__ float sm[256];
    sm[threadIdx.x] = ((int)threadIdx.x < n) ? partials[threadIdx.x] : 0.f;
    __syncthreads();
    for (int off = 128; off > 0; off >>= 1) {
        if ((int)threadIdx.x < off) sm[threadIdx.x] += sm[threadIdx.x + off];
        __syncthreads();
    }
    if (threadIdx.x == 0) out[0] = sm[0] * (1.0f / (float)NE);
}

extern "C" void kernel_launch(void* const* d_in, const int* in_sizes, int n_in,
                              void* d_out, int out_size, void* d_ws, size_t ws_size,
                              hipStream_t stream)
{
    const float* pred = (const float*)d_in[0];
    const float* targ = (const float*)d_in[1];
    float* m        = (float*)d_ws;       // NE floats   (1 MB, lives in L2)
    float* partials = m + NE;             // 256 floats

    pool_diff_kernel<<<NE / 256, 256, 0, stream>>>(pred, targ, m);

    // 256 blocks x 8 wave32 = 2048 waves; 2048*64 = 131072 divides NE exactly
    const int blocks2 = 256;
    grad_energy_wmma_kernel<<<blocks2, 256, 0, stream>>>(m, partials);

    finalize_kernel<<<1, 256, 0, stream>>>(partials, (float*)d_out, blocks2);
}